// FwFM_47021301957264
// MI455X (gfx1250) — compile-verified
//
#include <hip/hip_runtime.h>

// FwFM on MI455X (gfx1250, wave32).
//   out[b] = bias + sum_f lin[idx[b,f]] + 0.5 * sum( (W @ E_b) .* E_b )
// with W the symmetric 39x39 pair-weight matrix (zero diagonal), E_b the
// gathered 39x16 embedding block, both zero-padded to 48 so the GEMM runs as
// 3 Mtiles x 12 Ksteps of V_WMMA_F32_16X16X4_F32 (exact fp32 math).

#define FIELD   39
#define DIM     16
#define FPAD    48          // 39 padded to 3*16
#define WSTR    52          // LDS row stride for W (pad 48->52: conflict-free)
#define NCROSS  741         // 39*38/2
#define NWAVES  8
#define BLOCK   (NWAVES * 32)
#define BATCH   8192

typedef __attribute__((ext_vector_type(2))) float v2f;
typedef __attribute__((ext_vector_type(8))) float v8f;

__global__ __launch_bounds__(BLOCK)
void fwfm_wmma_kernel(const int*   __restrict__ inp,    // [BATCH, FIELD]
                      const float* __restrict__ emb,    // [1e6, DIM]
                      const float* __restrict__ fw,     // [NCROSS]
                      const float* __restrict__ lin,    // [1e6]
                      const float* __restrict__ bias_p, // scalar
                      float*       __restrict__ out)    // [BATCH]
{
    __shared__ float Wlds[FPAD * WSTR];              // ~9.8 KB, shared by block
    __shared__ float Elds[NWAVES][FPAD * DIM];       // 3 KB per wave

    const int tid  = threadIdx.x;
    const int wave = tid >> 5;
    const int lane = tid & 31;
    const int half = lane >> 4;      // which 16-lane half of the wave
    const int mn   = lane & 15;      // M for A-frag, N for B/C frags

    // ---- build symmetric pair-weight matrix W in LDS (once per block) ----
    for (int p = tid; p < FPAD * WSTR; p += BLOCK) Wlds[p] = 0.0f;
    __syncthreads();
    for (int p = tid; p < NCROSS; p += BLOCK) {
        int i = 0, rem = p;                         // pair p -> (i, j), i<j
        while (rem >= FIELD - 1 - i) { rem -= FIELD - 1 - i; ++i; }
        int j = i + 1 + rem;
        float w = fw[p];
        Wlds[i * WSTR + j] = w;
        Wlds[j * WSTR + i] = w;
    }

    // ---- per-wave: gather this sample's embedding block into LDS ----
    const int   b   = blockIdx.x * NWAVES + wave;   // grid exactly covers BATCH
    const int*  row = inp + (size_t)b * FIELD;
    float*      Ew  = Elds[wave];

    float lin_part = 0.0f;
    #pragma unroll
    for (int r = 0; r < FPAD / 2; ++r) {            // 2 rows/iter: 64B coalesced
        int   f = 2 * r + half;
        float v = 0.0f;
        if (f < FIELD) {
            int idx = row[f];
            v = emb[(size_t)idx * DIM + mn];
        }
        Ew[f * DIM + mn] = v;
    }
    // first-order gather: lane covers fields lane and lane+32
    if (lane < FIELD)      lin_part += lin[row[lane]];
    if (lane + 32 < FIELD) lin_part += lin[row[lane + 32]];

    __syncthreads();

    // ---- G = W @ E via V_WMMA_F32_16X16X4_F32; fold in G .* E on the fly ----
    float second = 0.0f;
    for (int mt = 0; mt < 3; ++mt) {
        v8f acc = {0.f, 0.f, 0.f, 0.f, 0.f, 0.f, 0.f, 0.f};
        #pragma unroll
        for (int kk = 0; kk < 12; ++kk) {
            const int kbase = kk * 4 + 2 * half;    // ISA 32-bit A/B layout
            v2f a, bb;
            a.x  = Wlds[(mt * 16 + mn) * WSTR + kbase];
            a.y  = Wlds[(mt * 16 + mn) * WSTR + kbase + 1];
            bb.x = Ew[(kbase)     * DIM + mn];
            bb.y = Ew[(kbase + 1) * DIM + mn];
            acc = __builtin_amdgcn_wmma_f32_16x16x4_f32(
                /*neg_a=*/false, a, /*neg_b=*/false, bb,
                /*c_mod=*/(short)0, acc, /*reuse_a=*/false, /*reuse_b=*/false);
        }
        // C/D layout: acc[v] holds G[M = mt*16 + v + 8*half][N = mn]
        #pragma unroll
        for (int v = 0; v < 8; ++v) {
            int M = mt * 16 + v + 8 * half;
            second += acc[v] * Ew[M * DIM + mn];
        }
    }

    // ---- wave reduction + output ----
    float total = lin_part + 0.5f * second;
    #pragma unroll
    for (int off = 16; off > 0; off >>= 1)
        total += __shfl_xor(total, off, 32);

    if (lane == 0)
        out[b] = total + bias_p[0];
}

extern "C" void kernel_launch(void* const* d_in, const int* in_sizes, int n_in,
                              void* d_out, int out_size, void* d_ws, size_t ws_size,
                              hipStream_t stream) {
    const int*   inp    = (const int*)  d_in[0];   // inputs            [8192,39] i32
    const float* emb    = (const float*)d_in[1];   // embedding_weights [1e6,16]  f32
    const float* fw     = (const float*)d_in[2];   // field_weights     [741,1]   f32
    const float* lin    = (const float*)d_in[3];   // linear_weights    [1e6]     f32
    const float* bias_p = (const float*)d_in[4];   // bias scalar
    float*       out    = (float*)d_out;           // [8192,1]

    dim3 grid(BATCH / NWAVES);   // 1024 blocks x 8 waves = one wave per sample
    dim3 block(BLOCK);
    fwfm_wmma_kernel<<<grid, block, 0, stream>>>(inp, emb, fw, lin, bias_p, out);
}